// RetNetAttention_82094004896541
// MI455X (gfx1250) — compile-verified
//
#include <hip/hip_runtime.h>
#include <hip/hip_bf16.h>
#include <math.h>

// ---------------------------------------------------------------------------
// RetNet attention, fp32, CDNA5 (gfx1250) WMMA path.
//   qkv = x @ W_qkv^T            (WMMA f32 16x16x4 GEMM, async-LDS staged)
//   k   = layernorm(k) per head  (wave32 shfl reductions)
//   y   = chunked retention scan (4x WMMA GEMMs per 64-chunk, async-LDS staged)
//   out = y @ W_o^T              (WMMA f32 16x16x4 GEMM)
// Data movement uses GLOBAL_LOAD_ASYNC_TO_LDS_B128 (ASYNCcnt) + double
// buffering so HBM latency overlaps the WMMA blocks.
// ---------------------------------------------------------------------------

typedef float v2f __attribute__((ext_vector_type(2)));
typedef float v8f __attribute__((ext_vector_type(8)));

__device__ __forceinline__ v8f wmma_f32(v2f a, v2f b, v8f c) {
  // 8 args: (neg_a, A, neg_b, B, c_mod, C, reuse_a, reuse_b)
  return __builtin_amdgcn_wmma_f32_16x16x4_f32(
      false, a, false, b, (short)0, c, false, false);
}

// Async copy 16B global -> LDS (per-lane), tracked by ASYNCcnt.
__device__ __forceinline__ void async_ld_b128(unsigned lds_addr, const float* g) {
  asm volatile("global_load_async_to_lds_b128 %0, %1, off"
               :: "v"(lds_addr), "v"(g) : "memory");
}
__device__ __forceinline__ void wait_async0() {
  asm volatile("s_wait_asynccnt 0" ::: "memory");
}
// Truncating a generic pointer to a __shared__ object yields the LDS offset.
__device__ __forceinline__ unsigned lds_off(const void* p) {
  return (unsigned)(size_t)p;
}

// ---------------------------------------------------------------------------
// Generic NT GEMM: C[M,N] = A[M,K] @ B[N,K]^T   (both row-major, K contiguous)
// M % 128 == 0, N % 64 == 0, K % 32 == 0 (true for all uses here).
// ---------------------------------------------------------------------------
#define TILE_M 128
#define TILE_N 64
#define TILE_K 32
#define KPAD   (TILE_K + 4)   // 36 floats = 144 B -> keeps 16B alignment

__global__ __launch_bounds__(256)
void gemm_nt_f32_wmma(const float* __restrict__ A, const float* __restrict__ B,
                      float* __restrict__ C, int M, int N, int K) {
  __shared__ float As[2][TILE_M][KPAD];
  __shared__ float Bs[2][TILE_N][KPAD];

  const int tid  = threadIdx.x;
  const int lane = tid & 31;
  const int wave = tid >> 5;          // 0..7
  const int half = lane >> 4;         // 0/1 -> K pair select
  const int lr   = lane & 15;
  const int m0   = blockIdx.y * TILE_M;
  const int n0   = blockIdx.x * TILE_N;
  const int rw   = wave * 16;         // wave's 16-row band

  v8f acc[4] = {};

  // Issue async stage for k0 = 0 into buffer 0.
  {
#pragma unroll
    for (int i = 0; i < 4; ++i) {
      int idx = tid + i * 256;
      int r = idx >> 3, c = (idx & 7) << 2;
      async_ld_b128(lds_off(&As[0][r][c]), A + (size_t)(m0 + r) * K + c);
    }
#pragma unroll
    for (int i = 0; i < 2; ++i) {
      int idx = tid + i * 256;
      int r = idx >> 3, c = (idx & 7) << 2;
      async_ld_b128(lds_off(&Bs[0][r][c]), B + (size_t)(n0 + r) * K + c);
    }
  }
  wait_async0();
  __syncthreads();

  int buf = 0;
  for (int k0 = 0; k0 < K; k0 += TILE_K) {
    // Prefetch next stage into the other buffer while we compute this one.
    if (k0 + TILE_K < K) {
      const int kn = k0 + TILE_K, nb = buf ^ 1;
#pragma unroll
      for (int i = 0; i < 4; ++i) {
        int idx = tid + i * 256;
        int r = idx >> 3, c = (idx & 7) << 2;
        async_ld_b128(lds_off(&As[nb][r][c]), A + (size_t)(m0 + r) * K + kn + c);
      }
#pragma unroll
      for (int i = 0; i < 2; ++i) {
        int idx = tid + i * 256;
        int r = idx >> 3, c = (idx & 7) << 2;
        async_ld_b128(lds_off(&Bs[nb][r][c]), B + (size_t)(n0 + r) * K + kn + c);
      }
    }

#pragma unroll
    for (int kk = 0; kk < TILE_K; kk += 4) {
      v2f a;
      a.x = As[buf][rw + lr][kk + 2 * half];
      a.y = As[buf][rw + lr][kk + 2 * half + 1];
#pragma unroll
      for (int t = 0; t < 4; ++t) {
        v2f b;
        b.x = Bs[buf][t * 16 + lr][kk + 2 * half];
        b.y = Bs[buf][t * 16 + lr][kk + 2 * half + 1];
        acc[t] = wmma_f32(a, b, acc[t]);
      }
    }
    wait_async0();
    __syncthreads();
    buf ^= 1;
  }

  // C layout: vgpr e -> row e (lanes 0-15) / row e+8 (lanes 16-31); col = lr.
#pragma unroll
  for (int t = 0; t < 4; ++t)
#pragma unroll
    for (int e = 0; e < 8; ++e)
      C[(size_t)(m0 + rw + e + 8 * half) * N + (n0 + t * 16 + lr)] = acc[t][e];
}

// ---------------------------------------------------------------------------
// LayerNorm over dk=64 of the K part of qkv[m, 1024 + h*64 .. +64].
// One wave per (m,h) row; 2 elements per lane.
// ---------------------------------------------------------------------------
__global__ __launch_bounds__(256)
void k_layernorm_kernel(float* __restrict__ qkv, int nrows) {
  int wid  = (blockIdx.x * 256 + threadIdx.x) >> 5;
  int lane = threadIdx.x & 31;
  if (wid >= nrows) return;
  int m = wid >> 4;       // row / H
  int h = wid & 15;       // row % H
  float* p = qkv + (size_t)m * 3072 + 1024 + h * 64;
  float x0 = p[lane], x1 = p[lane + 32];
  float s = x0 + x1, sq = x0 * x0 + x1 * x1;
#pragma unroll
  for (int off = 16; off; off >>= 1) {
    s  += __shfl_xor(s, off, 32);
    sq += __shfl_xor(sq, off, 32);
  }
  float mu  = s * (1.0f / 64.0f);
  float var = sq * (1.0f / 64.0f) - mu * mu;
  float r   = rsqrtf(var + 1e-5f);
  p[lane]      = (x0 - mu) * r;
  p[lane + 32] = (x1 - mu) * r;
}

// ---------------------------------------------------------------------------
// Chunked retention. One block per (b,h); 4 waves; chunk C = 64, dk = 64.
//   P      = (Q K^T) .* decay_mask                      -> Am (LDS)
//   Y      = diag(g^{i+1}) * (Q S^T) + Am V             -> y (global)
//   S_new  = g^C * S + (diag(g^{C-1-i}) V)^T K          -> Ss (LDS)
// Q/K/V chunk tiles are double-buffered with async-to-LDS so the next
// chunk's 48 KB streams in during this chunk's four 64^3 WMMA GEMMs.
// ---------------------------------------------------------------------------
#define CHUNK 64
#define DK    64
#define LPAD  68   // 64 + 4 skew; 272 B row stride keeps 16B alignment

__global__ __launch_bounds__(128)
void retention_chunked(const float* __restrict__ qkv,
                       const float* __restrict__ log_gamma,
                       float* __restrict__ y, int L) {
  __shared__ float Qs[2][CHUNK][LPAD];
  __shared__ float Ks[2][CHUNK][LPAD];
  __shared__ float Vs[2][CHUNK][LPAD];
  __shared__ float Ss[DK][LPAD];
  __shared__ float Am[CHUNK][LPAD];
  __shared__ float gp[CHUNK + 1];

  const int b    = blockIdx.x >> 4;   // / H
  const int h    = blockIdx.x & 15;   // % H
  const int tid  = threadIdx.x;
  const int lane = tid & 31;
  const int wave = tid >> 5;          // 0..3
  const int half = lane >> 4;
  const int lr   = lane & 15;
  const int rw   = wave * 16;

  const size_t bh_base = ((size_t)(b * L)) * 3072 + h * 64;

  // Issue async stage of chunk 0 into buffer 0.
#pragma unroll
  for (int i = 0; i < 8; ++i) {
    int idx = tid + i * 128;            // 0..1023
    int r = idx >> 4, c = (idx & 15) << 2;
    const float* src = qkv + bh_base + (size_t)r * 3072;
    async_ld_b128(lds_off(&Qs[0][r][c]), src + c);
    async_ld_b128(lds_off(&Ks[0][r][c]), src + 1024 + c);
    async_ld_b128(lds_off(&Vs[0][r][c]), src + 2048 + c);
  }

  const float gamma = 1.0f / (1.0f + __expf(-log_gamma[h]));
  for (int i = tid; i <= CHUNK; i += 128) gp[i] = powf(gamma, (float)i);
  float* sp = &Ss[0][0];
  for (int i = tid; i < DK * LPAD; i += 128) sp[i] = 0.0f;
  wait_async0();
  __syncthreads();
  const float gC = gp[CHUNK];

  int buf = 0;
  for (int c0 = 0; c0 < L; c0 += CHUNK) {
    // ---- prefetch next chunk's Q,K,V into the other buffer ----------------
    if (c0 + CHUNK < L) {
      const int nb = buf ^ 1;
      const size_t base = bh_base + (size_t)(c0 + CHUNK) * 3072;
#pragma unroll
      for (int i = 0; i < 8; ++i) {
        int idx = tid + i * 128;
        int r = idx >> 4, c = (idx & 15) << 2;
        const float* src = qkv + base + (size_t)r * 3072;
        async_ld_b128(lds_off(&Qs[nb][r][c]), src + c);
        async_ld_b128(lds_off(&Ks[nb][r][c]), src + 1024 + c);
        async_ld_b128(lds_off(&Vs[nb][r][c]), src + 2048 + c);
      }
    }

    // ---- P = Q K^T, apply causal decay mask, store to Am ------------------
    {
      v8f p[4] = {};
#pragma unroll
      for (int kk = 0; kk < DK; kk += 4) {
        v2f a;
        a.x = Qs[buf][rw + lr][kk + 2 * half];
        a.y = Qs[buf][rw + lr][kk + 2 * half + 1];
#pragma unroll
        for (int t = 0; t < 4; ++t) {
          v2f bb;
          bb.x = Ks[buf][t * 16 + lr][kk + 2 * half];
          bb.y = Ks[buf][t * 16 + lr][kk + 2 * half + 1];
          p[t] = wmma_f32(a, bb, p[t]);
        }
      }
#pragma unroll
      for (int t = 0; t < 4; ++t)
#pragma unroll
        for (int e = 0; e < 8; ++e) {
          int i = rw + e + 8 * half, j = t * 16 + lr;
          float f = (j <= i) ? gp[i - j] : 0.0f;
          Am[i][j] = p[t][e] * f;
        }
    }
    __syncthreads();

    // ---- Y = diag(g^{i+1}) (Q S^T) + Am V ---------------------------------
    v8f yacc[4] = {};
#pragma unroll
    for (int kk = 0; kk < DK; kk += 4) {       // kk = e (state col)
      v2f a;
      a.x = Qs[buf][rw + lr][kk + 2 * half];
      a.y = Qs[buf][rw + lr][kk + 2 * half + 1];
#pragma unroll
      for (int t = 0; t < 4; ++t) {            // output col d = t*16+lr
        v2f bb;                                // B[e][d] = S[d][e]
        bb.x = Ss[t * 16 + lr][kk + 2 * half];
        bb.y = Ss[t * 16 + lr][kk + 2 * half + 1];
        yacc[t] = wmma_f32(a, bb, yacc[t]);
      }
    }
#pragma unroll
    for (int t = 0; t < 4; ++t)
#pragma unroll
      for (int e = 0; e < 8; ++e)
        yacc[t][e] *= gp[(rw + e + 8 * half) + 1];

#pragma unroll
    for (int kk = 0; kk < CHUNK; kk += 4) {    // kk = j (chunk time)
      v2f a;
      a.x = Am[rw + lr][kk + 2 * half];
      a.y = Am[rw + lr][kk + 2 * half + 1];
#pragma unroll
      for (int t = 0; t < 4; ++t) {            // B[j][d] = V[j][d]
        v2f bb;
        bb.x = Vs[buf][kk + 2 * half][t * 16 + lr];
        bb.y = Vs[buf][kk + 2 * half + 1][t * 16 + lr];
        yacc[t] = wmma_f32(a, bb, yacc[t]);
      }
    }
#pragma unroll
    for (int t = 0; t < 4; ++t)
#pragma unroll
      for (int e = 0; e < 8; ++e) {
        int i = rw + e + 8 * half, j = t * 16 + lr;
        y[(size_t)(b * L + c0 + i) * 1024 + h * 64 + j] = yacc[t][e];
      }

    // ---- S_new = g^C S + (diag(g^{C-1-i}) V)^T K --------------------------
    v8f sacc[4];
#pragma unroll
    for (int t = 0; t < 4; ++t)
#pragma unroll
      for (int e = 0; e < 8; ++e)
        sacc[t][e] = gC * Ss[rw + e + 8 * half][t * 16 + lr];

#pragma unroll
    for (int kk = 0; kk < CHUNK; kk += 4) {    // kk = i (chunk time)
      v2f a;                                   // A[d][i] = w_i * V[i][d], d = rw+lr
      a.x = gp[CHUNK - 1 - (kk + 2 * half)]     * Vs[buf][kk + 2 * half][rw + lr];
      a.y = gp[CHUNK - 1 - (kk + 2 * half + 1)] * Vs[buf][kk + 2 * half + 1][rw + lr];
#pragma unroll
      for (int t = 0; t < 4; ++t) {            // B[i][e] = K[i][e]
        v2f bb;
        bb.x = Ks[buf][kk + 2 * half][t * 16 + lr];
        bb.y = Ks[buf][kk + 2 * half + 1][t * 16 + lr];
        sacc[t] = wmma_f32(a, bb, sacc[t]);
      }
    }
    __syncthreads();   // everyone done reading Ss / Am / current Q,K,V
#pragma unroll
    for (int t = 0; t < 4; ++t)
#pragma unroll
      for (int e = 0; e < 8; ++e)
        Ss[rw + e + 8 * half][t * 16 + lr] = sacc[t][e];
    wait_async0();     // next chunk's async tiles have landed
    __syncthreads();   // Ss + next buffers ready for next chunk
    buf ^= 1;
  }
}

// ---------------------------------------------------------------------------
extern "C" void kernel_launch(void* const* d_in, const int* in_sizes, int n_in,
                              void* d_out, int out_size, void* d_ws, size_t ws_size,
                              hipStream_t stream) {
  const float* x     = (const float*)d_in[0];   // [2,4096,1024]
  const float* W_qkv = (const float*)d_in[1];   // [3072,1024]
  const float* W_o   = (const float*)d_in[2];   // [1024,1024]
  const float* lg    = (const float*)d_in[3];   // [16]
  float* out = (float*)d_out;                   // [2,4096,1024]

  const int B = 2, L = 4096, D = 1024;
  const int M = B * L;                          // 8192

  float* qkv = (float*)d_ws;                    // M x 3072  (96 MB)
  float* yws = qkv + (size_t)M * 3 * D;         // M x 1024  (32 MB)

  // 1) qkv = x @ W_qkv^T
  gemm_nt_f32_wmma<<<dim3((3 * D) / TILE_N, M / TILE_M), 256, 0, stream>>>(
      x, W_qkv, qkv, M, 3 * D, D);

  // 2) layernorm K per head
  k_layernorm_kernel<<<(M * 16) / 8, 256, 0, stream>>>(qkv, M * 16);

  // 3) chunked retention scan
  retention_chunked<<<B * 16, 128, 0, stream>>>(qkv, lg, yws, L);

  // 4) out = y @ W_o^T
  gemm_nt_f32_wmma<<<dim3(D / TILE_N, M / TILE_M), 256, 0, stream>>>(
      yws, W_o, out, M, D, D);
}